// StructureExtractor_4587025072633
// MI455X (gfx1250) — compile-verified
//
#include <hip/hip_runtime.h>
#include <hip/hip_bf16.h>

// GIN (2 layers): agg = h + segment_sum(h[src] -> dst); h' = relu(agg @ W + b)
// N_NODES=100000, N_EDGES=1600000, D=64.
//
// Roofline: scatter = 102.4M f32 atomics + ~1.7GB L2 traffic per layer
// (dominant); GEMM = 0.82 GFLOP per layer (~us at 23.3 TB/s, negligible) ->
// keep exact f32 semantics via V_WMMA_F32_16X16X4_F32 instead of bf16.

#define N_NODES 100000
#define N_EDGES 1600000
#define D 64

typedef __attribute__((ext_vector_type(2))) float v2f;
typedef __attribute__((ext_vector_type(8))) float v8f;

// ---------------------------------------------------------------------------
// Kernel 1: agg[i] = h[i]   (folds the (1+eps)*h self term, eps == 0)
// ---------------------------------------------------------------------------
__global__ __launch_bounds__(256) void gin_init_agg(const float4* __restrict__ h4,
                                                    float4* __restrict__ agg4,
                                                    int n4) {
  int i = blockIdx.x * 256 + threadIdx.x;
  if (i < n4) agg4[i] = h4[i];
}

// ---------------------------------------------------------------------------
// Kernel 2: edge scatter-add.  16 threads per edge; each thread moves 4
// consecutive f32 features with one b128 gather + 4 hardware f32 atomics.
// Inline asm guarantees global_atomic_add_f32 (no-return, STOREcnt) at
// device scope — no CAS fallback.
// ---------------------------------------------------------------------------
__device__ __forceinline__ void atomic_add_f32_dev(float* p, float v) {
  asm volatile("global_atomic_add_f32 %0, %1, off scope:SCOPE_DEV"
               :
               : "v"(p), "v"(v)
               : "memory");
}

__global__ __launch_bounds__(256) void gin_scatter(const float* __restrict__ h,
                                                   const int* __restrict__ src,
                                                   const int* __restrict__ dst,
                                                   float* __restrict__ agg,
                                                   int n_edges) {
  int t = blockIdx.x * 256 + threadIdx.x;
  int e = t >> 4;
  if (e >= n_edges) return;
  int f = (t & 15) << 2;  // feature base: 0,4,...,60
  int s = src[e];
  int d = dst[e];
  const float4 v = *(const float4*)(h + s * D + f);
  float* p = agg + d * D + f;
  atomic_add_f32_dev(p + 0, v.x);
  atomic_add_f32_dev(p + 1, v.y);
  atomic_add_f32_dev(p + 2, v.z);
  atomic_add_f32_dev(p + 3, v.w);
}

// ---------------------------------------------------------------------------
// Kernel 3: out = relu(A @ W + bias), A: [n_nodes, 64] f32, W: [64, 64] f32.
// One wave per 32x16 output block = TWO 16x16 WMMA tiles sharing B fragments:
//  - two independent accumulator chains (hides WMMA latency within the wave)
//  - halves the strided scalar W loads (each B fragment feeds 2 WMMAs)
// K=64 via 16 steps of V_WMMA_F32_16X16X4_F32 per chain (full f32 precision).
//
// Fragment layouts (ISA 7.12.2, 32-bit):
//   A 16x4 : lane l(0..15) VGPR{0,1} = A[m0+l][4k+{0,1}];
//            lanes 16..31 hold K+2  -> contiguous pair => v2f (b64) load.
//   B 4x16 : lane l VGPR{0,1} = W[4k+2*half+{0,1}][n0+l]
//   C 16x16: VGPR v, lane l: M = v + 8*half, N = l
// ---------------------------------------------------------------------------
__global__ __launch_bounds__(256) void gin_gemm_bias_relu(
    const float* __restrict__ A, const float* __restrict__ W,
    const float* __restrict__ bias, float* __restrict__ out, int n_nodes) {
  const int lane = threadIdx.x & 31;
  const int wave = threadIdx.x >> 5;
  const int tile = blockIdx.x * 8 + wave;
  const int mpairs = n_nodes >> 5;          // 3125 pairs of M-tiles
  const int tileM = tile >> 2;              // 4 N-tiles across D=64
  const int tileN = tile & 3;
  if (tileM >= mpairs) return;              // wave-uniform guard (EXEC stays all-ones)
  const int m0 = tileM << 5;                // 32 rows per wave
  const int n0 = tileN << 4;
  const int half = lane >> 4;               // 0: K+{0,1}, 1: K+{2,3}
  const int l = lane & 15;

  const float* arow0 = A + (m0 + l) * D + 2 * half;      // contiguous K pair
  const float* arow1 = arow0 + 16 * D;                   // second M tile
  const float* wcol = W + (2 * half) * D + n0 + l;       // stride-D K pair

  v8f c0 = {};
  v8f c1 = {};
#pragma unroll
  for (int k = 0; k < 16; ++k) {
    v2f b;
    b.x = wcol[(4 * k + 0) * D];
    b.y = wcol[(4 * k + 1) * D];
    v2f a0 = *(const v2f*)(arow0 + 4 * k);               // global_load_b64
    v2f a1 = *(const v2f*)(arow1 + 4 * k);
    // 8 args: (neg_a, A, neg_b, B, c_mod, C, reuse_a, reuse_b)
    c0 = __builtin_amdgcn_wmma_f32_16x16x4_f32(false, a0, false, b, (short)0,
                                               c0, false, false);
    c1 = __builtin_amdgcn_wmma_f32_16x16x4_f32(false, a1, false, b, (short)0,
                                               c1, false, false);
  }

  const float bj = bias[n0 + l];
  float* ocol = out + n0 + l;
#pragma unroll
  for (int v = 0; v < 8; ++v) {
    float x0 = c0[v] + bj;
    float x1 = c1[v] + bj;
    ocol[(m0 + v + 8 * half) * D] = x0 > 0.0f ? x0 : 0.0f;
    ocol[(m0 + 16 + v + 8 * half) * D] = x1 > 0.0f ? x1 : 0.0f;
  }
}

// ---------------------------------------------------------------------------
// Host-side orchestration (all on `stream`, graph-capture safe).
// Inputs: x, src, dst, W1, b1, W2, b2.  d_ws holds agg (25.6 MB).
// Layer 1: agg=x;  scatter(x)   -> agg; h1  = gemm(agg,W1,b1) -> d_out
// Layer 2: agg=h1; scatter(h1)  -> agg; out = gemm(agg,W2,b2) -> d_out
// Every scratch byte is rewritten each call => deterministic across replays.
// ---------------------------------------------------------------------------
extern "C" void kernel_launch(void* const* d_in, const int* in_sizes, int n_in,
                              void* d_out, int out_size, void* d_ws,
                              size_t ws_size, hipStream_t stream) {
  const float* x = (const float*)d_in[0];
  const int* src = (const int*)d_in[1];
  const int* dst = (const int*)d_in[2];
  const float* W1 = (const float*)d_in[3];
  const float* b1 = (const float*)d_in[4];
  const float* W2 = (const float*)d_in[5];
  const float* b2 = (const float*)d_in[6];
  float* out = (float*)d_out;
  float* agg = (float*)d_ws;  // N_NODES * D floats

  const int n4 = N_NODES * D / 4;                        // 1.6M float4
  const int init_blocks = (n4 + 255) / 256;
  const int scat_blocks = (N_EDGES * 16 + 255) / 256;    // 16 threads/edge
  const int gemm_waves = (N_NODES / 32) * 4;             // 12500
  const int gemm_blocks = (gemm_waves + 7) / 8;          // 1563

  // ---- Layer 1 ----
  gin_init_agg<<<init_blocks, 256, 0, stream>>>((const float4*)x, (float4*)agg, n4);
  gin_scatter<<<scat_blocks, 256, 0, stream>>>(x, src, dst, agg, N_EDGES);
  gin_gemm_bias_relu<<<gemm_blocks, 256, 0, stream>>>(agg, W1, b1, out, N_NODES);

  // ---- Layer 2 ----
  gin_init_agg<<<init_blocks, 256, 0, stream>>>((const float4*)out, (float4*)agg, n4);
  gin_scatter<<<scat_blocks, 256, 0, stream>>>(out, src, dst, agg, N_EDGES);
  gin_gemm_bias_relu<<<gemm_blocks, 256, 0, stream>>>(agg, W2, b2, out, N_NODES);
}